// CTCLoss_56478819942961
// MI455X (gfx1250) — compile-verified
//
#include <hip/hip_runtime.h>
#include <math.h>

// Problem constants (from reference setup_inputs): B=64, T=2048, C=1024, S=256
#define B_    64
#define T_    2048
#define C_    1024
#define S_    256
#define L_    513          // 2S+1 extended states
#define EWP   260          // padded emit row stride in floats (1040 B, 16B-aligned)
#define NEGV  (-1e30f)

// ---------------------------------------------------------------------------
// CDNA5 async global->LDS copy (16 bytes per lane) + async counter wait.
// The clang builtin exists but is declared with CUDA __device__/__shared__
// language address spaces that cannot be spelled from HIP source, so emit the
// instruction directly per CDNA5 ISA:
//   global_load_async_to_lds_b128 vDST(LDS byte addr), vADDR(64-bit), off
// Generic LDS pointers are {aperture_hi32, lds_offset_lo32}; the low 32 bits
// are exactly the LDS byte address the instruction's VDST expects.
// ---------------------------------------------------------------------------
__device__ __forceinline__ void async_ld16(void* lds_dst, const void* gsrc) {
  unsigned l = (unsigned)(unsigned long long)lds_dst;  // LDS byte address
  asm volatile("global_load_async_to_lds_b128 %0, %1, off"
               :: "v"(l), "v"(gsrc) : "memory");
}

__device__ __forceinline__ void wait_async0() {
#if __has_builtin(__builtin_amdgcn_s_wait_asynccnt)
  __builtin_amdgcn_s_wait_asynccnt(0);
#else
  asm volatile("s_wait_asynccnt 0" ::: "memory");
#endif
}

// logaddexp, safe for the -1e30 "log zero" surrogate (matches jnp.logaddexp
// to fp32 rounding: -1e30 + log(2) rounds back to -1e30).
__device__ __forceinline__ float lae(float a, float b) {
  float mx = fmaxf(a, b);
  float mn = fminf(a, b);
  return mx + log1pf(expf(mn - mx));
}

// ---------------------------------------------------------------------------
// Phase 1: per-(b,t) row -> log_softmax normalizer + compact emit row.
// emit[row*EWP + 0]      = lp[b,t,0]            (blank, normalized)
// emit[row*EWP + 1 + s]  = lp[b,t,targets[b,s]] (label s, normalized)
// One block per row; row streamed into LDS via async B128 loads.
// ---------------------------------------------------------------------------
__global__ __launch_bounds__(256) void ctc_lse_gather_kernel(
    const float* __restrict__ lp, const int* __restrict__ targets,
    float* __restrict__ emit)
{
  __shared__ __align__(16) float row[C_];
  __shared__ float red[8];

  const int      tid = threadIdx.x;
  const unsigned rid = blockIdx.x;        // rid = b*T + t
  const int      b   = rid >> 11;         // T = 2^11

  const float* g = lp + (size_t)rid * C_;

  // async copy 4KB row into LDS: 256 lanes x 16B
  async_ld16(&row[tid * 4], g + tid * 4);
  wait_async0();
  __syncthreads();

  const float4 v = *reinterpret_cast<const float4*>(&row[tid * 4]);

  // block max
  float m = fmaxf(fmaxf(v.x, v.y), fmaxf(v.z, v.w));
  #pragma unroll
  for (int o = 16; o >= 1; o >>= 1) m = fmaxf(m, __shfl_xor(m, o, 32));
  if ((tid & 31) == 0) red[tid >> 5] = m;
  __syncthreads();
  float bm = red[0];
  #pragma unroll
  for (int i = 1; i < 8; ++i) bm = fmaxf(bm, red[i]);

  // block sum of exp(x - max)
  float s = expf(v.x - bm) + expf(v.y - bm) + expf(v.z - bm) + expf(v.w - bm);
  #pragma unroll
  for (int o = 16; o >= 1; o >>= 1) s += __shfl_xor(s, o, 32);
  __syncthreads();                        // everyone done reading red
  if ((tid & 31) == 0) red[tid >> 5] = s;
  __syncthreads();
  float bs = red[0];
  #pragma unroll
  for (int i = 1; i < 8; ++i) bs += red[i];

  const float lse = bm + logf(bs);

  // compact gather: blank + S labels (labels guaranteed in [1,C))
  const size_t base = (size_t)rid * EWP;
  if (tid < S_) {
    const int cls = targets[b * S_ + tid];
    emit[base + 1 + tid] = row[cls] - lse;
  }
  if (tid == 0) emit[base] = row[0] - lse;
}

// ---------------------------------------------------------------------------
// Phase 2: alpha recursion. One block per batch element, 288 threads (9 waves).
// Thread i owns states 2i (blank) and 2i+1 (label s=i); i=256 owns only 512.
// Alpha double-buffered in LDS as float2 pairs with a left halo slot.
// Emit rows double-buffered in LDS, fetched with async B128 loads overlapped
// with compute (one s_wait_asynccnt + one barrier per time step).
// ---------------------------------------------------------------------------
__global__ __launch_bounds__(288) void ctc_alpha_kernel(
    const float* __restrict__ emit, const int* __restrict__ targets,
    const int* __restrict__ in_len, const int* __restrict__ tgt_len,
    float* __restrict__ scores)
{
  __shared__ __align__(16) float2 abuf[2][S_ + 2];   // [0]=halo, [1+i]=pair i
  __shared__ __align__(16) float  ebuf[2][EWP];

  const int tid = threadIdx.x;
  const int b   = blockIdx.x;
  const int len = in_len[b];     // in [T/2, T]
  const int tl  = tgt_len[b];    // in [S/2, S]

  const float* erow = emit + (size_t)b * T_ * EWP;

  // can_skip for odd state 2i+1 (s=i): i>=1 and targets differ (ext != 0 always)
  bool skip = false;
  if (tid >= 1 && tid < S_)
    skip = targets[b * S_ + tid] != targets[b * S_ + tid - 1];

  // prologue: emit row 0 -> ebuf[0]
  if (tid < 65) async_ld16(&ebuf[0][tid * 4], erow + tid * 4);
  wait_async0();
  __syncthreads();

  // alpha at t=0: only states 0 and 1 start at 0 + emit
  float a_e = NEGV, a_o = NEGV;
  if (tid == 0) { a_e = ebuf[0][0]; a_o = ebuf[0][1]; }
  if (tid <= S_) abuf[0][tid + 1] = make_float2(a_e, a_o);
  if (tid == 0) {                               // permanent halo (alpha[-2],alpha[-1])
    abuf[0][0] = make_float2(NEGV, NEGV);
    abuf[1][0] = make_float2(NEGV, NEGV);
  }
  // kick off row 1 (len >= T/2 >= 2 always here)
  if (tid < 65) async_ld16(&ebuf[1][tid * 4], erow + (size_t)EWP + tid * 4);

  int p = 0;
  for (int t = 1; t < len; ++t) {
    wait_async0();          // emit row t landed in LDS (my wave's loads)
    __syncthreads();        // ...and everyone else's; alpha[t-1] stores visible

    // prefetch row t+1 into the buffer that held row t-1 (safe past barrier)
    if (t + 1 < len && tid < 65)
      async_ld16(&ebuf[(t + 1) & 1][tid * 4],
                 erow + (size_t)(t + 1) * EWP + tid * 4);

    const float2* ac = abuf[p];
    const float*  ec = ebuf[t & 1];

    if (tid <= S_) {
      const float2 left = ac[tid];     // (alpha[2i-2], alpha[2i-1]) at t-1
      const float  e0   = ec[0];       // blank emit (all even states)

      // even state 2i: paths from 2i and 2i-1 (no skip into blank)
      const float ne = lae(a_e, left.y) + e0;

      // odd state 2i+1: paths from 2i+1, 2i, and (skip) 2i-1
      float no = NEGV;
      if (tid < S_) {
        float acc = lae(a_o, a_e);
        if (skip) acc = lae(acc, left.y);
        no = acc + ec[1 + tid];
      }
      a_e = ne; a_o = no;
      abuf[p ^ 1][tid + 1] = make_float2(ne, no);
    }
    p ^= 1;
  }
  __syncthreads();

  if (tid == 0) {
    const int i1 = 2 * tl;                   // <= 512
    const int i2 = (2 * tl - 1 > 0) ? (2 * tl - 1) : 0;
    const float2 p1 = abuf[p][(i1 >> 1) + 1];
    const float2 p2 = abuf[p][(i2 >> 1) + 1];
    const float a1 = (i1 & 1) ? p1.y : p1.x;
    float       a2 = (i2 & 1) ? p2.y : p2.x;
    if (tl <= 0) a2 = NEGV;
    scores[b] = lae(a1, a2);
  }
}

// ---------------------------------------------------------------------------
// Phase 3: -mean of 64 per-batch log-likelihoods (single wave32).
// ---------------------------------------------------------------------------
__global__ __launch_bounds__(32) void ctc_finalize_kernel(
    const float* __restrict__ scores, float* __restrict__ out)
{
  float v = scores[threadIdx.x] + scores[threadIdx.x + 32];
  #pragma unroll
  for (int o = 16; o >= 1; o >>= 1) v += __shfl_xor(v, o, 32);
  if (threadIdx.x == 0) out[0] = -v / (float)B_;
}

// ---------------------------------------------------------------------------
// Host launcher.
// ws layout: [ emit: B*T*EWP floats (~130 MB) | scores: B floats ]
// ---------------------------------------------------------------------------
extern "C" void kernel_launch(void* const* d_in, const int* in_sizes, int n_in,
                              void* d_out, int out_size, void* d_ws, size_t ws_size,
                              hipStream_t stream) {
  (void)in_sizes; (void)n_in; (void)out_size; (void)ws_size;

  const float* lp      = (const float*)d_in[0];   // [B,T,C] fp32
  const int*   targets = (const int*)d_in[1];     // [B,S]
  const int*   in_len  = (const int*)d_in[2];     // [B]
  const int*   tgt_len = (const int*)d_in[3];     // [B]

  float* emit   = (float*)d_ws;
  float* scores = emit + (size_t)B_ * T_ * EWP;

  ctc_lse_gather_kernel<<<B_ * T_, 256, 0, stream>>>(lp, targets, emit);
  ctc_alpha_kernel<<<B_, 288, 0, stream>>>(emit, targets, in_len, tgt_len, scores);
  ctc_finalize_kernel<<<1, 32, 0, stream>>>(scores, (float*)d_out);
}